// GATConv_22084721836670
// MI455X (gfx1250) — compile-verified
//
#include <hip/hip_runtime.h>
#include <math.h>

#define N_NODES   50000
#define N_EDGES   800000
#define IN_DIM    32
#define OUT_DIM   32
#define NUM_HEADS 4
#define HP_DIM    128   // OUT_DIM * NUM_HEADS
#define NEG_SLOPE 0.01f

typedef __attribute__((ext_vector_type(2))) float v2f;
typedef __attribute__((ext_vector_type(8))) float v8f;

// ---------------------------------------------------------------------------
// Kernel 1: hp = h @ W_lin^T + b_lin  via V_WMMA_F32_16X16X4_F32
//   h: (50000,32) row-major, W: (128,32) row-major (so B-tile = W^T slice)
//   One wave per 16-node M-tile; 8 N-tiles x 8 K-steps = 64 WMMAs per wave.
// ---------------------------------------------------------------------------
__global__ void k_gemm_hp(const float* __restrict__ h,
                          const float* __restrict__ W,
                          const float* __restrict__ bias,
                          float* __restrict__ hp) {
  const int wave = (int)((blockIdx.x * blockDim.x + threadIdx.x) >> 5);
  const int lane = (int)(threadIdx.x & 31);
  const int m0   = wave * 16;
  if (m0 >= N_NODES) return;              // wave-uniform: EXEC stays all-ones
  const int lm  = lane & 15;              // A: row in tile; B/C: column in tile
  const int lhi = lane >> 4;              // K-pair select (A/B), M+8 select (C/D)
  const int row = m0 + lm;

  // A fragments for all 8 K-steps: a[ks] covers K = ks*4 .. ks*4+3
  v2f a[8];
#pragma unroll
  for (int ks = 0; ks < 8; ++ks) {
    const float* p = h + row * IN_DIM + ks * 4 + 2 * lhi;
    a[ks].x = p[0];                        // K = ks*4 + 2*lhi
    a[ks].y = p[1];                        // K = ks*4 + 2*lhi + 1
  }

#pragma unroll
  for (int nt = 0; nt < 8; ++nt) {
    v8f c = {};
#pragma unroll
    for (int ks = 0; ks < 8; ++ks) {
      // B[k][n] = W[(n0+n)*32 + k], same K striping as A
      const float* q = W + (nt * 16 + lm) * IN_DIM + ks * 4 + 2 * lhi;
      v2f b; b.x = q[0]; b.y = q[1];
      c = __builtin_amdgcn_wmma_f32_16x16x4_f32(
              false, a[ks], false, b, (short)0, c, false, false);
    }
    const float bv = bias[nt * 16 + lm];
#pragma unroll
    for (int v = 0; v < 8; ++v) {
      // C/D layout: M = m0 + v + 8*lhi, N = nt*16 + lm
      hp[(m0 + v + 8 * lhi) * HP_DIM + nt * 16 + lm] = c[v] + bv;
    }
  }
}

// ---------------------------------------------------------------------------
// Kernel 2: per-(node,head) attention scores s_src/s_dst (dot over 32 dims)
// ---------------------------------------------------------------------------
__global__ void k_scores(const float* __restrict__ hp,
                         const float* __restrict__ Wa,   // (4, 64)
                         float* __restrict__ s_src,
                         float* __restrict__ s_dst) {
  const int t = blockIdx.x * blockDim.x + threadIdx.x;
  if (t >= N_NODES * NUM_HEADS) return;
  const int n = t >> 2, hh = t & 3;
  const float* hr   = hp + (size_t)n * HP_DIM + hh * OUT_DIM;
  const float* wsrc = Wa + hh * 2 * OUT_DIM;
  const float* wdst = wsrc + OUT_DIM;
  float ss = 0.f, sd = 0.f;
#pragma unroll
  for (int d = 0; d < OUT_DIM; ++d) {
    const float v = hr[d];
    ss += v * wsrc[d];
    sd += v * wdst[d];
  }
  s_src[t] = ss;
  s_dst[t] = sd;
}

// ---------------------------------------------------------------------------
// CSR construction: histogram of dst, exclusive scan, scatter edge sources.
// ---------------------------------------------------------------------------
__global__ void k_hist(const int* __restrict__ dst, int* __restrict__ deg) {
  const int i = blockIdx.x * blockDim.x + threadIdx.x;
  if (i < N_EDGES) atomicAdd(&deg[dst[i]], 1);
}

__global__ void k_scan(const int* __restrict__ deg, int* __restrict__ rowptr) {
  __shared__ int sh[1024];
  __shared__ int runsh;
  const int tid = threadIdx.x;
  if (tid == 0) { runsh = 0; rowptr[0] = 0; }
  __syncthreads();
  for (int base = 0; base < N_NODES; base += 1024) {
    const int i = base + tid;
    sh[tid] = (i < N_NODES) ? deg[i] : 0;
    __syncthreads();
    for (int off = 1; off < 1024; off <<= 1) {
      const int t = (tid >= off) ? sh[tid - off] : 0;
      __syncthreads();
      sh[tid] += t;
      __syncthreads();
    }
    const int inc = sh[tid] + runsh;       // inclusive prefix + running offset
    if (i < N_NODES) rowptr[i + 1] = inc;
    __syncthreads();
    if (tid == 1023) runsh = inc;
    __syncthreads();
  }
}

__global__ void k_scatter(const int* __restrict__ src,
                          const int* __restrict__ dst,
                          const int* __restrict__ rowptr,
                          int* __restrict__ cursor,
                          int* __restrict__ csr_src) {
  const int i = blockIdx.x * blockDim.x + threadIdx.x;
  if (i >= N_EDGES) return;
  const int d = dst[i];
  const int p = atomicAdd(&cursor[d], 1);
  csr_src[rowptr[d] + p] = src[i];
}

// ---------------------------------------------------------------------------
// Kernel 3: per-(node,head) online-softmax aggregation. One wave32 per
// (node, head); lane d owns output dim d. Zero floating-point atomics.
// ---------------------------------------------------------------------------
__global__ void k_gat(const float* __restrict__ hp,
                      const float* __restrict__ s_src,
                      const float* __restrict__ s_dst,
                      const float* __restrict__ b_att,
                      const int* __restrict__ rowptr,
                      const int* __restrict__ csr_src,
                      float* __restrict__ out) {
  const int gw   = (int)((blockIdx.x * blockDim.x + threadIdx.x) >> 5);
  const int lane = (int)(threadIdx.x & 31);
  if (gw >= N_NODES * NUM_HEADS) return;
  const int n  = gw >> 2;
  const int hh = gw & 3;
  const int beg = rowptr[n], end = rowptr[n + 1];
  const float sd = s_dst[n * NUM_HEADS + hh] + b_att[hh];

  float m = -INFINITY, z = 0.f, acc = 0.f;
  for (int k = beg; k < end; ++k) {
    const int s = csr_src[k];
    float e = s_src[s * NUM_HEADS + hh] + sd;
    e = (e > 0.f) ? e : NEG_SLOPE * e;          // leaky_relu
    const float mn = fmaxf(m, e);
    const float sc = __expf(m - mn);            // first iter: exp(-inf) = 0
    const float p  = __expf(e - mn);
    z   = z * sc + p;
    acc = acc * sc + p * hp[(size_t)s * HP_DIM + hh * OUT_DIM + lane];
    m = mn;
  }
  out[(size_t)n * HP_DIM + hh * OUT_DIM + lane] = (end > beg) ? acc / z : 0.f;
}

// ---------------------------------------------------------------------------
extern "C" void kernel_launch(void* const* d_in, const int* in_sizes, int n_in,
                              void* d_out, int out_size, void* d_ws, size_t ws_size,
                              hipStream_t stream) {
  (void)in_sizes; (void)n_in; (void)out_size; (void)ws_size;
  const float* h_in  = (const float*)d_in[0];
  const float* W_lin = (const float*)d_in[1];
  const float* b_lin = (const float*)d_in[2];
  const float* W_att = (const float*)d_in[3];
  const float* b_att = (const float*)d_in[4];
  const int*   src   = (const int*)d_in[5];
  const int*   dst   = (const int*)d_in[6];
  float* out = (float*)d_out;

  // Workspace layout (~31 MB total)
  float* hp    = (float*)d_ws;                             // 6,400,000 f32
  float* s_src = hp + (size_t)N_NODES * HP_DIM;            //   200,000 f32
  float* s_dst = s_src + N_NODES * NUM_HEADS;              //   200,000 f32
  int*   deg   = (int*)(s_dst + N_NODES * NUM_HEADS);      //    50,000 i32
  int*   cur   = deg + N_NODES;                            //    50,000 i32
  int*   rowp  = cur + N_NODES;                            //    50,001 i32 (pad 50,004)
  int*   csrc  = rowp + 50004;                             //   800,000 i32

  hipMemsetAsync(deg, 0, N_NODES * sizeof(int), stream);
  hipMemsetAsync(cur, 0, N_NODES * sizeof(int), stream);

  // 1) GEMM: 3125 waves of 16 nodes each, 128-thread blocks (4 waves)
  {
    const int waves = N_NODES / 16;                        // 3125
    const int threads = waves * 32;
    k_gemm_hp<<<(threads + 127) / 128, 128, 0, stream>>>(h_in, W_lin, b_lin, hp);
  }
  // 2) attention scores
  k_scores<<<(N_NODES * NUM_HEADS + 255) / 256, 256, 0, stream>>>(hp, W_att, s_src, s_dst);
  // 3) CSR build
  k_hist<<<(N_EDGES + 255) / 256, 256, 0, stream>>>(dst, deg);
  k_scan<<<1, 1024, 0, stream>>>(deg, rowp);
  k_scatter<<<(N_EDGES + 255) / 256, 256, 0, stream>>>(src, dst, rowp, cur, csrc);
  // 4) online-softmax aggregation: one wave per (node, head)
  {
    const long long threads = (long long)N_NODES * NUM_HEADS * 32;   // 6.4M
    k_gat<<<(int)((threads + 255) / 256), 256, 0, stream>>>(
        hp, s_src, s_dst, b_att, rowp, csrc, out);
  }
}